// FourierSelfAttention_20796231647312
// MI455X (gfx1250) — compile-verified
//
#include <hip/hip_runtime.h>
#include <math.h>

typedef __attribute__((ext_vector_type(2))) float v2f;
typedef __attribute__((ext_vector_type(8))) float v8f;

static constexpr int kB   = 4;
static constexpr int kT   = 4096;
static constexpr int kC   = 1024;
static constexpr int kH   = 16;
static constexpr int kDH  = 64;
static constexpr int kM   = kB * kT;          // 16384 rows
static constexpr int kF   = kT / 2 + 1;       // 2049 frequency bins
static constexpr int kFP  = 2064;             // padded
static constexpr size_t kCHUNK = (size_t)kB * kH * kDH * kT;  // 16777216 floats = 64 MB

// ---------------------------------------------------------------------------
// CDNA5 async global->LDS copy of one contiguous 16 KB row (4096 floats),
// 256 threads, 16 B per lane-op, tracked with ASYNCcnt.
// ---------------------------------------------------------------------------
__device__ inline void async_row_to_lds(const float* __restrict__ g, float* lds_dst, int tid)
{
    const unsigned lbase = (unsigned)(size_t)(void*)lds_dst;   // LDS byte offset (addr[31:0])
    #pragma unroll
    for (int k = 0; k < 4; ++k) {
        const int ci = tid + k * 256;          // 16-byte chunk index 0..1023
        const float* ga = g + ci * 4;
        const unsigned la = lbase + ci * 16;
        asm volatile("global_load_async_to_lds_b128 %0, %1, off"
                     :: "v"(la), "v"(ga) : "memory");
    }
}
__device__ inline void wait_async_lds()
{
    asm volatile("s_wait_asynccnt 0" ::: "memory");
}

// ---------------------------------------------------------------------------
// In-place radix-2 complex FFT, N=4096, 256 threads, split re/im in LDS.
// sign = -1 forward (numpy convention), +1 inverse (caller scales by 1/N).
// ---------------------------------------------------------------------------
__device__ inline void fft4096_soa(float* re, float* im, int tid, float sign)
{
    // bit-reversal permutation (12 bits); disjoint pairs, owner = smaller index
    for (int i = tid; i < 4096; i += 256) {
        const int j = (int)(__brev((unsigned)i) >> 20);
        if (j > i) {
            float tr = re[i], ti = im[i];
            re[i] = re[j]; im[i] = im[j];
            re[j] = tr;    im[j] = ti;
        }
    }
    __syncthreads();
    for (int s = 1; s <= 12; ++s) {
        const int half = 1 << (s - 1);
        const float ang = sign * 6.28318530717958647692f / (float)(1 << s);
        for (int idx = tid; idx < 2048; idx += 256) {
            const int pos = idx & (half - 1);
            const int i0  = ((idx >> (s - 1)) << s) + pos;
            const int i1  = i0 + half;
            float sn, cs;
            __sincosf(ang * (float)pos, &sn, &cs);
            const float ur = re[i0], ui = im[i0];
            const float wr = re[i1], wi = im[i1];
            const float tr = cs * wr - sn * wi;
            const float ti = cs * wi + sn * wr;
            re[i0] = ur + tr; im[i0] = ui + ti;
            re[i1] = ur - tr; im[i1] = ui - ti;
        }
        __syncthreads();
    }
}

// ---------------------------------------------------------------------------
// K1/K3: qkv projection. Out[m, c0+nloc] = sum_k X[m,k] * W_qkv[c0+nloc, k]
// stored TRANSPOSED per channel: Outt[sel*CHUNK + ((b*H+h)*DH+d)*T + t],
// sel = nloc/1024 (consecutive 64MB chunks), c = (c0+nloc)%1024, h=c/64, d=c%64.
// Block = 8 waves, tile 128x64. Wave tile 16x64 via 4x V_WMMA_F32_16X16X4_F32.
// ---------------------------------------------------------------------------
__global__ __launch_bounds__(256) void qkv_gemm_wmma(
    const float* __restrict__ X, const float* __restrict__ Wq,
    float* __restrict__ Outt, int c0)
{
    constexpr int K = kC;
    const int lane  = threadIdx.x & 31;
    const int wave  = threadIdx.x >> 5;
    const int mBase = blockIdx.x * 128 + wave * 16;
    const int nBase = blockIdx.y * 64;
    const int kg2   = (lane >> 4) << 1;   // 0 or 2: K sub-pair for A/B fragments

    const float* arow  = X  + (size_t)(mBase + (lane & 15)) * K + kg2;
    const float* brow0 = Wq + (size_t)(c0 + nBase + (lane & 15)) * K + kg2;
    const float* brow1 = brow0 + (size_t)16 * K;
    const float* brow2 = brow0 + (size_t)32 * K;
    const float* brow3 = brow0 + (size_t)48 * K;

    v8f acc[4] = {};
    #pragma unroll 4
    for (int k = 0; k < K; k += 4) {
        v2f a  = *(const v2f*)(arow  + k);
        v2f b0 = *(const v2f*)(brow0 + k);
        v2f b1 = *(const v2f*)(brow1 + k);
        v2f b2 = *(const v2f*)(brow2 + k);
        v2f b3 = *(const v2f*)(brow3 + k);
        acc[0] = __builtin_amdgcn_wmma_f32_16x16x4_f32(false, a, false, b0, (short)0, acc[0], false, false);
        acc[1] = __builtin_amdgcn_wmma_f32_16x16x4_f32(false, a, false, b1, (short)0, acc[1], false, false);
        acc[2] = __builtin_amdgcn_wmma_f32_16x16x4_f32(false, a, false, b2, (short)0, acc[2], false, false);
        acc[3] = __builtin_amdgcn_wmma_f32_16x16x4_f32(false, a, false, b3, (short)0, acc[3], false, false);
    }

    // store transposed: per lane, rows r=0..7 are 8 consecutive t values
    const int b  = mBase >> 12;                       // row / T (tiles never straddle)
    const int t0 = (mBase & (kT - 1)) + ((lane >> 4) << 3);
    #pragma unroll
    for (int j = 0; j < 4; ++j) {
        const int cl  = nBase + j * 16 + (lane & 15); // local column in this pass
        const int sel = cl >> 10;                     // which of Q/K/V chunk
        const int c   = (c0 + cl) & (kC - 1);
        float* dst = Outt + (size_t)sel * kCHUNK
                   + ((size_t)((b * kH + (c >> 6)) * kDH + (c & 63))) * kT + t0;
        *(float4*)(dst)     = make_float4(acc[j][0], acc[j][1], acc[j][2], acc[j][3]);
        *(float4*)(dst + 4) = make_float4(acc[j][4], acc[j][5], acc[j][6], acc[j][7]);
    }
}

// ---------------------------------------------------------------------------
// K2: per (b,h): FFT q and k rows for each d, accumulate Re(conj(q̂)k̂),
// softmax over F bins. LDS ~58.6 KB: re/im FFT buffers + cached q̂ bins.
// Rows arrive via GLOBAL_LOAD_ASYNC_TO_LDS_B128 (ASYNCcnt).
// ---------------------------------------------------------------------------
__global__ __launch_bounds__(256) void spectral_scores_softmax(
    const float* __restrict__ Qt, const float* __restrict__ Kt,
    float* __restrict__ attn)
{
    extern __shared__ float smem[];
    float* reB = smem;                 // 4096
    float* imB = reB + 4096;           // 4096
    float* qhr = imB + 4096;           // kFP
    float* qhi = qhr + kFP;            // kFP
    float* sc  = qhi + kFP;            // kFP
    float* red = sc  + kFP;            // 256

    const int tid = threadIdx.x;
    const int bh  = blockIdx.x;
    for (int f = tid; f < kFP; f += 256) sc[f] = 0.f;
    __syncthreads();

    const float* qb = Qt + (size_t)bh * kDH * kT;
    const float* kb = Kt + (size_t)bh * kDH * kT;

    for (int d = 0; d < kDH; ++d) {
        // ---- q row: async copy reals to LDS, zero imaginary meanwhile ----
        async_row_to_lds(qb + (size_t)d * kT, reB, tid);
        for (int i = tid; i < kT; i += 256) imB[i] = 0.f;
        wait_async_lds();
        __syncthreads();
        fft4096_soa(reB, imB, tid, -1.f);
        for (int f = tid; f < kF; f += 256) { qhr[f] = reB[f]; qhi[f] = imB[f]; }
        __syncthreads();

        // ---- k row ----
        async_row_to_lds(kb + (size_t)d * kT, reB, tid);
        for (int i = tid; i < kT; i += 256) imB[i] = 0.f;
        wait_async_lds();
        __syncthreads();
        fft4096_soa(reB, imB, tid, -1.f);
        for (int f = tid; f < kF; f += 256)
            sc[f] += qhr[f] * reB[f] + qhi[f] * imB[f];  // Re(conj(q)k)
        __syncthreads();
    }

    // softmax over f = 0..2048, scale = dh^-0.5 = 0.125
    const float scale = 0.125f;
    float lm = -3.4e38f;
    for (int f = tid; f < kF; f += 256) lm = fmaxf(lm, sc[f] * scale);
    red[tid] = lm; __syncthreads();
    for (int s = 128; s > 0; s >>= 1) { if (tid < s) red[tid] = fmaxf(red[tid], red[tid + s]); __syncthreads(); }
    const float mx = red[0];
    __syncthreads();
    float ls = 0.f;
    for (int f = tid; f < kF; f += 256) { float e = __expf(sc[f] * scale - mx); sc[f] = e; ls += e; }
    red[tid] = ls; __syncthreads();
    for (int s = 128; s > 0; s >>= 1) { if (tid < s) red[tid] += red[tid + s]; __syncthreads(); }
    const float inv = 1.f / red[0];
    for (int f = tid; f < kF; f += 256) attn[(size_t)bh * kFP + f] = sc[f] * inv;
}

// ---------------------------------------------------------------------------
// K4: per (b,h,d): FFT(v), multiply by symmetric real gate attn[min(f,T-f)],
// inverse FFT, write real part to out_mid[B,T,C] at channel h*64+d.
// ---------------------------------------------------------------------------
__global__ __launch_bounds__(256) void spectral_filter_irfft(
    const float* __restrict__ Vt, const float* __restrict__ attn,
    float* __restrict__ out_mid)
{
    extern __shared__ float smem[];
    float* reB = smem;                 // 4096
    float* imB = reB + 4096;           // 4096
    float* aw  = imB + 4096;           // kFP

    const int tid = threadIdx.x;
    const int row = blockIdx.x;            // (b*H + h)*DH + d
    const int bh  = row / kDH;
    const int d   = row - bh * kDH;

    async_row_to_lds(Vt + (size_t)row * kT, reB, tid);
    for (int i = tid; i < kT; i += 256) imB[i] = 0.f;
    for (int f = tid; f < kF; f += 256) aw[f] = attn[(size_t)bh * kFP + f];
    wait_async_lds();
    __syncthreads();

    fft4096_soa(reB, imB, tid, -1.f);
    for (int i = tid; i < kT; i += 256) {
        const int f = (i <= kT / 2) ? i : (kT - i);   // Hermitian extension
        const float m = aw[f];
        reB[i] *= m; imB[i] *= m;
    }
    __syncthreads();
    fft4096_soa(reB, imB, tid, 1.f);

    const int b = bh / kH, h = bh - (bh / kH) * kH;
    float* dst = out_mid + (size_t)b * kT * kC + (size_t)(h * kDH + d);
    const float invN = 1.f / 4096.f;
    for (int t = tid; t < kT; t += 256) dst[(size_t)t * kC] = reB[t] * invN;
}

// ---------------------------------------------------------------------------
// K5: output projection: Y[m,n] = sum_k X[m,k]*W_out[n,k] + bias[n]
// ---------------------------------------------------------------------------
__global__ __launch_bounds__(256) void out_gemm_wmma(
    const float* __restrict__ X, const float* __restrict__ Wo,
    const float* __restrict__ bias, float* __restrict__ Y)
{
    constexpr int K = kC, N = kC;
    const int lane  = threadIdx.x & 31;
    const int wave  = threadIdx.x >> 5;
    const int mBase = blockIdx.x * 128 + wave * 16;
    const int nBase = blockIdx.y * 64;
    const int kg2   = (lane >> 4) << 1;

    const float* arow  = X  + (size_t)(mBase + (lane & 15)) * K + kg2;
    const float* brow0 = Wo + (size_t)(nBase + (lane & 15)) * K + kg2;
    const float* brow1 = brow0 + (size_t)16 * K;
    const float* brow2 = brow0 + (size_t)32 * K;
    const float* brow3 = brow0 + (size_t)48 * K;

    v8f acc[4] = {};
    #pragma unroll 4
    for (int k = 0; k < K; k += 4) {
        v2f a  = *(const v2f*)(arow  + k);
        v2f b0 = *(const v2f*)(brow0 + k);
        v2f b1 = *(const v2f*)(brow1 + k);
        v2f b2 = *(const v2f*)(brow2 + k);
        v2f b3 = *(const v2f*)(brow3 + k);
        acc[0] = __builtin_amdgcn_wmma_f32_16x16x4_f32(false, a, false, b0, (short)0, acc[0], false, false);
        acc[1] = __builtin_amdgcn_wmma_f32_16x16x4_f32(false, a, false, b1, (short)0, acc[1], false, false);
        acc[2] = __builtin_amdgcn_wmma_f32_16x16x4_f32(false, a, false, b2, (short)0, acc[2], false, false);
        acc[3] = __builtin_amdgcn_wmma_f32_16x16x4_f32(false, a, false, b3, (short)0, acc[3], false, false);
    }

    const int m0 = mBase + ((lane >> 4) << 3);
    #pragma unroll
    for (int j = 0; j < 4; ++j) {
        const int n  = nBase + j * 16 + (lane & 15);
        const float bn = bias[n];
        #pragma unroll
        for (int r = 0; r < 8; ++r)
            Y[(size_t)(m0 + r) * N + n] = acc[j][r] + bn;
    }
}

// ---------------------------------------------------------------------------
extern "C" void kernel_launch(void* const* d_in, const int* in_sizes, int n_in,
                              void* d_out, int out_size, void* d_ws, size_t ws_size,
                              hipStream_t stream)
{
    (void)in_sizes; (void)n_in; (void)out_size; (void)ws_size;
    const float* x     = (const float*)d_in[0];
    const float* W_qkv = (const float*)d_in[1];
    const float* W_out = (const float*)d_in[2];
    const float* b_out = (const float*)d_in[3];
    float* out = (float*)d_out;
    float* ws  = (float*)d_ws;

    // ws layout (floats): [reg0: CHUNK][reg1: CHUNK][attn: 64*kFP]
    // reg0 holds Q, later reused for V; reg1 holds K, later reused for mid.
    float* reg0 = ws;
    float* reg1 = ws + kCHUNK;
    float* attn = ws + 2 * kCHUNK;

    dim3 blk(256);

    // 1) Q,K projection (W_qkv cols 0..2047) -> reg0 (Q) + reg1 (K), transposed
    qkv_gemm_wmma<<<dim3(kM / 128, 2048 / 64), blk, 0, stream>>>(x, W_qkv, reg0, 0);

    // 2) spectral scores + softmax -> attn [64, kFP]
    const size_t smemB = (size_t)(4096 * 2 + kFP * 3 + 256) * sizeof(float); // ~58.6 KB
    spectral_scores_softmax<<<dim3(kB * kH), blk, smemB, stream>>>(reg0, reg1, attn);

    // 3) V projection (W_qkv cols 2048..3071) -> reg0 (V), transposed
    qkv_gemm_wmma<<<dim3(kM / 128, 1024 / 64), blk, 0, stream>>>(x, W_qkv, reg0, 2048);

    // 4) gate in frequency domain + inverse FFT -> reg1 as [B,T,C]
    const size_t smemC = (size_t)(4096 * 2 + kFP) * sizeof(float);           // ~41 KB
    spectral_filter_irfft<<<dim3(kB * kH * kDH), blk, smemC, stream>>>(reg0, attn, reg1);

    // 5) output projection + bias -> d_out
    out_gemm_wmma<<<dim3(kM / 128, 1024 / 64), blk, 0, stream>>>(reg1, W_out, b_out, out);
}